// UCVLADPRunner_60464549593849
// MI455X (gfx1250) — compile-verified
//
#include <hip/hip_runtime.h>
#include <math.h>

// ---------------- problem constants (fixed by setup_inputs) ----------------
#define BB   256     // batch
#define TT   512     // time
#define AA   64      // action dim
#define DZ   256     // latent dim
#define UU   1024    // user rows
#define SS   127     // num sdtw windows: (512-8)/4 + 1
#define WW   8       // sdtw window
#define STR  4       // sdtw stride
#define MARGIN_F 0.5f
#define BIG_F 1000000000.0f

// ---------------- workspace layout (float index offsets) -------------------
#define MSE_P    0                       // 2048 block partials
#define KL_P     2048                    // 256 block partials
#define TRIP_P   2304                    // 256 per-b hinge values
#define ORTHO_P  2560                    // 4096 per-tile partials
#define SDTW_P   6656                    // 256 per-b hinge values
#define WN_OFF   8192                    // 1024*256 normalized W (32KB-aligned)
#define POSD_OFF (WN_OFF + UU * DZ)      // 256*127 sdtw-pos results
#define NEGD_OFF (POSD_OFF + BB * SS)    // 256*127 sdtw-neg results

typedef __attribute__((ext_vector_type(2))) float v2f;
typedef __attribute__((ext_vector_type(8))) float v8f;

// ============================ MSE partials ================================
// 2048 blocks x 256 threads; 4096 floats (1024 float4) per block.
// float4 loads -> global_load_b128: 4x fewer VMEM issues on the one kernel
// that is HBM-bandwidth dominant (2 x 33.5MB streamed once).
__global__ void mse_kernel(const float* __restrict__ pred,
                           const float* __restrict__ target,
                           float* __restrict__ ws) {
    __shared__ float red[256];
    const float4* p4 = (const float4*)pred;
    const float4* t4 = (const float4*)target;
    size_t base = (size_t)blockIdx.x * 1024 + threadIdx.x;
    float s = 0.0f;
#pragma unroll
    for (int i = 0; i < 4; ++i) {
        size_t idx = base + (size_t)i * 256;
        float4 a = p4[idx];
        float4 b = t4[idx];
        float d0 = a.x - b.x, d1 = a.y - b.y, d2 = a.z - b.z, d3 = a.w - b.w;
        s += d0 * d0 + d1 * d1 + d2 * d2 + d3 * d3;
    }
    red[threadIdx.x] = s;
    __syncthreads();
    for (int off = 128; off > 0; off >>= 1) {
        if ((int)threadIdx.x < off) red[threadIdx.x] += red[threadIdx.x + off];
        __syncthreads();
    }
    if (threadIdx.x == 0) ws[MSE_P + blockIdx.x] = red[0];
}

// ============================ KL partials =================================
// 256 blocks x 256 threads, one element each (B*Dz = 65536).
__global__ void kl_kernel(const float* __restrict__ mu,
                          const float* __restrict__ log_var,
                          float* __restrict__ ws) {
    __shared__ float red[256];
    int idx = blockIdx.x * 256 + threadIdx.x;
    float m = mu[idx], lv = log_var[idx];
    float v = -0.5f * (1.0f + lv - m * m - __expf(lv));
    red[threadIdx.x] = v;
    __syncthreads();
    for (int off = 128; off > 0; off >>= 1) {
        if ((int)threadIdx.x < off) red[threadIdx.x] += red[threadIdx.x + off];
        __syncthreads();
    }
    if (threadIdx.x == 0) ws[KL_P + blockIdx.x] = red[0];
}

// ============================ Triplet per-b ===============================
// 32 blocks x 256 threads = 256 waves; wave per batch element.
// 2 float4 loads per lane per row (Dz=256 -> 64 float4 / 32 lanes).
__global__ void triplet_kernel(const float* __restrict__ z,
                               const long long* __restrict__ pos_idx,
                               const long long* __restrict__ neg_idx,
                               const float* __restrict__ valid,
                               float* __restrict__ ws) {
    int lane = threadIdx.x & 31;
    int b = blockIdx.x * 8 + (threadIdx.x >> 5);
    int p = (int)pos_idx[b];
    int n = (int)neg_idx[b];
    const float4* zb = (const float4*)(z + (size_t)b * DZ);
    const float4* zp = (const float4*)(z + (size_t)p * DZ);
    const float4* zn = (const float4*)(z + (size_t)n * DZ);
    float da = 0.0f, dn = 0.0f;
    const float eps = 1e-6f;
#pragma unroll
    for (int i = 0; i < 2; ++i) {
        int d = lane + i * 32;
        float4 a = zb[d], q = zp[d], r = zn[d];
        float ax = a.x - q.x + eps, ay = a.y - q.y + eps,
              az = a.z - q.z + eps, aw = a.w - q.w + eps;
        da += ax * ax + ay * ay + az * az + aw * aw;
        float bx = a.x - r.x + eps, by = a.y - r.y + eps,
              bz = a.z - r.z + eps, bw = a.w - r.w + eps;
        dn += bx * bx + by * by + bz * bz + bw * bw;
    }
    for (int off = 16; off > 0; off >>= 1) {
        da += __shfl_xor(da, off, 32);
        dn += __shfl_xor(dn, off, 32);
    }
    if (lane == 0) {
        float h = valid[b] * fmaxf(sqrtf(da) - sqrtf(dn) + MARGIN_F, 0.0f);
        ws[TRIP_P + b] = h;
    }
}

// ===================== Row-normalize W into workspace =====================
// 128 blocks x 256 threads = 1024 waves; wave per row; float4 I/O.
__global__ void rownorm_kernel(const float* __restrict__ W,
                               float* __restrict__ ws) {
    int lane = threadIdx.x & 31;
    int u = blockIdx.x * 8 + (threadIdx.x >> 5);
    const float4* row = (const float4*)(W + (size_t)u * DZ);
    float4 v[2];
    float s = 0.0f;
#pragma unroll
    for (int i = 0; i < 2; ++i) {
        v[i] = row[lane + i * 32];
        s += v[i].x * v[i].x + v[i].y * v[i].y + v[i].z * v[i].z +
             v[i].w * v[i].w;
    }
    for (int off = 16; off > 0; off >>= 1) s += __shfl_xor(s, off, 32);
    float inv = 1.0f / sqrtf(s);
    float4* out = (float4*)(ws + WN_OFF + (size_t)u * DZ);
#pragma unroll
    for (int i = 0; i < 2; ++i) {
        float4 o;
        o.x = v[i].x * inv; o.y = v[i].y * inv;
        o.z = v[i].z * inv; o.w = v[i].w * inv;
        out[lane + i * 32] = o;
    }
}

// ==================== Ortho: Gram via f32 WMMA 16x16x4 ====================
// 512 blocks x 256 threads = 4096 waves; wave per 16x16 Gram tile.
// A-frag (16x4 f32): lane L: M = L&15, k-half = 2*(L>>4), 2 VGPRs = K,K+1.
// B-frag (4x16 f32): lane L: N = L&15, same k-half layout (B = Wn^T rows).
// C/D (16x16 f32): vgpr v, lanes 0-15: (M=v, N=lane); lanes 16-31: (M=v+8).
__global__ void ortho_kernel(const float* __restrict__ wsWn,
                             float* __restrict__ ws) {
    int lane = threadIdx.x & 31;
    int wid = blockIdx.x * 8 + (threadIdx.x >> 5);   // 0..4095
    int i0 = (wid >> 6) * 16;
    int j0 = (wid & 63) * 16;
    int m = lane & 15;
    int khalf = (lane >> 4) * 2;
    const float* Arow = wsWn + (size_t)(i0 + m) * DZ + khalf;
    const float* Brow = wsWn + (size_t)(j0 + m) * DZ + khalf;  // col n = lane&15
    v8f c = {};
    for (int k = 0; k < DZ; k += 4) {
        v2f a, b;
        a.x = Arow[k];  a.y = Arow[k + 1];
        b.x = Brow[k];  b.y = Brow[k + 1];
        c = __builtin_amdgcn_wmma_f32_16x16x4_f32(false, a, false, b,
                                                  (short)0, c, false, false);
    }
    int nn = lane & 15;
    int mbase = (lane >> 4) * 8;
    float s = 0.0f;
#pragma unroll
    for (int v = 0; v < 8; ++v) {
        int gi = i0 + mbase + v;
        int gj = j0 + nn;
        float g = c[v] - ((gi == gj) ? 1.0f : 0.0f);
        s += g * g;
    }
    for (int off = 16; off > 0; off >>= 1) s += __shfl_xor(s, off, 32);
    if (lane == 0) ws[ORTHO_P + wid] = s;
}

// ================== Soft-DTW windows: WMMA cross-terms + DP ===============
// 4064 blocks x 256 threads = 32512 waves; wave per (b, s) window.
// One 16x16 f32 WMMA accumulation (K=64) computes BOTH pos (N=0..7) and neg
// (N=8..15) cross dot-product matrices. Lanes 0/1 then run the two 8x8
// soft-DTW DPs (softmin3 via min-shifted logsumexp, gamma = 1).
// pred (33.5MB) fits in the 192MB L2, so all window re-reads are L2 hits.
__global__ void sdtw_kernel(const float* __restrict__ pred,
                            const long long* __restrict__ pos_idx,
                            const long long* __restrict__ neg_idx,
                            float* __restrict__ ws) {
    __shared__ float cross[8][8][16];   // [warp][i][col]  col<8: pos, col>=8: neg
    __shared__ float norms[8][24];      // [warp][row]  0-7: a, 8-15: p, 16-23: n

    int lane = threadIdx.x & 31;
    int warp = threadIdx.x >> 5;
    int id = blockIdx.x * 8 + warp;     // 0..32511 exactly
    int b = id / SS;
    int s = id % SS;
    int st = s * STR;
    int p = (int)pos_idx[b];
    int n = (int)neg_idx[b];

    // --- WMMA cross terms: A rows = window(b) (dup for M=8..15),
    //     B cols 0-7 = window(p), cols 8-15 = window(n) ---
    int m = lane & 15;
    int khalf = (lane >> 4) * 2;
    int arow = st + (m & 7);                       // duplicate for m>=8
    int nsrc = (m < 8) ? p : n;
    int brow = st + (m & 7);
    const float* pa = pred + ((size_t)b * TT + arow) * AA + khalf;
    const float* pb = pred + ((size_t)nsrc * TT + brow) * AA + khalf;
    v8f c = {};
    for (int k = 0; k < AA; k += 4) {
        v2f a, bfrag;
        a.x = pa[k];      a.y = pa[k + 1];
        bfrag.x = pb[k];  bfrag.y = pb[k + 1];
        c = __builtin_amdgcn_wmma_f32_16x16x4_f32(false, a, false, bfrag,
                                                  (short)0, c, false, false);
    }
    if (lane < 16) {
#pragma unroll
        for (int v = 0; v < 8; ++v) cross[warp][v][lane] = c[v];
    }

    // --- squared norms of the 24 distinct rows (one lane per row) ---
    // rows are 64 floats = 16 float4 (256B-aligned) -> b128 loads.
    if (lane < 24) {
        int src = (lane < 8) ? b : ((lane < 16) ? p : n);
        int row = st + (lane & 7);
        const float4* q = (const float4*)(pred + ((size_t)src * TT + row) * AA);
        float sum = 0.0f;
#pragma unroll
        for (int d = 0; d < AA / 4; ++d) {
            float4 v = q[d];
            sum += v.x * v.x + v.y * v.y + v.z * v.z + v.w * v.w;
        }
        norms[warp][lane] = sum;
    }
    // single-wave producer/consumer on LDS: compiler inserts s_wait_dscnt;
    // cross-wave slices are disjoint, so no block barrier needed.

    // --- soft-DTW DP on lanes 0 (pos) and 1 (neg) ---
    if (lane < 2) {
        int which = lane;                // 0 = pos, 1 = neg
        float prev[WW + 1], cur[WW + 1];
        prev[0] = 0.0f;
#pragma unroll
        for (int j = 1; j <= WW; ++j) prev[j] = BIG_F;
        for (int i = 1; i <= WW; ++i) {
            cur[0] = BIG_F;
            float na = norms[warp][i - 1];
#pragma unroll
            for (int j = 1; j <= WW; ++j) {
                float nb = norms[warp][8 + which * 8 + (j - 1)];
                float cr = cross[warp][i - 1][(j - 1) + which * 8];
                float cost = fmaxf(na + nb - 2.0f * cr, 0.0f);
                float dd = prev[j - 1], uu = prev[j], ll = cur[j - 1];
                float mn = fminf(dd, fminf(uu, ll));
                // softmin3 (gamma=1): mn - log(sum exp(mn - x))
                float sm = mn - __logf(__expf(mn - dd) + __expf(mn - uu) +
                                       __expf(mn - ll));
                cur[j] = cost + sm;
            }
#pragma unroll
            for (int j = 0; j <= WW; ++j) prev[j] = cur[j];
        }
        float* outbuf = ws + (which == 0 ? POSD_OFF : NEGD_OFF);
        outbuf[id] = prev[WW];
    }
}

// ================= Soft-DTW per-b min/max + hinge =========================
// 32 blocks x 256 threads = 256 waves; wave per batch element.
__global__ void sdtw_reduce_kernel(const float* __restrict__ valid,
                                   float* __restrict__ ws) {
    int lane = threadIdx.x & 31;
    int b = blockIdx.x * 8 + (threadIdx.x >> 5);
    const float* dp = ws + POSD_OFF + (size_t)b * SS;
    const float* dn = ws + NEGD_OFF + (size_t)b * SS;
    float mn = 3.4e38f, mx = -3.4e38f;
    for (int s = lane; s < SS; s += 32) {
        mn = fminf(mn, dp[s]);
        mx = fmaxf(mx, dn[s]);
    }
    for (int off = 16; off > 0; off >>= 1) {
        mn = fminf(mn, __shfl_xor(mn, off, 32));
        mx = fmaxf(mx, __shfl_xor(mx, off, 32));
    }
    if (lane == 0)
        ws[SDTW_P + b] = valid[b] * fmaxf(mn - mx + MARGIN_F, 0.0f);
}

// =========================== Finalize =====================================
// One block, deterministic fixed-order strided sums + LDS tree reduction.
__device__ float block_strided_sum(const float* __restrict__ p, int n,
                                   float* red) {
    float s = 0.0f;
    for (int i = threadIdx.x; i < n; i += 256) s += p[i];
    red[threadIdx.x] = s;
    __syncthreads();
    for (int off = 128; off > 0; off >>= 1) {
        if ((int)threadIdx.x < off) red[threadIdx.x] += red[threadIdx.x + off];
        __syncthreads();
    }
    float r = red[0];
    __syncthreads();
    return r;
}

__global__ void finalize_kernel(const float* __restrict__ valid,
                                const float* __restrict__ ws,
                                float* __restrict__ out) {
    __shared__ float red[256];
    float mse_s   = block_strided_sum(ws + MSE_P,   2048, red);
    float kl_s    = block_strided_sum(ws + KL_P,     256, red);
    float trip_s  = block_strided_sum(ws + TRIP_P,   256, red);
    float ortho_s = block_strided_sum(ws + ORTHO_P, 4096, red);
    float sdtw_s  = block_strided_sum(ws + SDTW_P,   256, red);
    float valid_s = block_strided_sum(valid,         256, red);
    if (threadIdx.x == 0) {
        float cnt   = fmaxf(valid_s, 1.0f);
        float mse   = mse_s / (float)((size_t)BB * TT * AA);
        float kl    = kl_s / (float)(BB * DZ);
        float trip  = trip_s / cnt;
        float sdtw  = sdtw_s / cnt;
        float ortho = sqrtf(ortho_s);
        out[0] = mse + 0.001f * kl + 0.1f * trip + 0.1f * sdtw + 0.01f * ortho;
    }
}

// ============================ launch ======================================
extern "C" void kernel_launch(void* const* d_in, const int* in_sizes, int n_in,
                              void* d_out, int out_size, void* d_ws,
                              size_t ws_size, hipStream_t stream) {
    (void)in_sizes; (void)n_in; (void)out_size; (void)ws_size;
    const float*     pred    = (const float*)d_in[0];
    const float*     target  = (const float*)d_in[1];
    const float*     z       = (const float*)d_in[2];
    const float*     mu      = (const float*)d_in[3];
    const float*     log_var = (const float*)d_in[4];
    const float*     W       = (const float*)d_in[5];
    // d_in[6] = user_id (unused; pos/neg precomputed by harness)
    const long long* pos_idx = (const long long*)d_in[7];
    const long long* neg_idx = (const long long*)d_in[8];
    const float*     valid   = (const float*)d_in[9];
    float* out = (float*)d_out;
    float* ws  = (float*)d_ws;

    mse_kernel     <<<2048, 256, 0, stream>>>(pred, target, ws);
    kl_kernel      <<< 256, 256, 0, stream>>>(mu, log_var, ws);
    triplet_kernel <<<  32, 256, 0, stream>>>(z, pos_idx, neg_idx, valid, ws);
    rownorm_kernel <<< 128, 256, 0, stream>>>(W, ws);
    ortho_kernel   <<< 512, 256, 0, stream>>>(ws + WN_OFF, ws);
    sdtw_kernel    <<<4064, 256, 0, stream>>>(pred, pos_idx, neg_idx, ws);
    sdtw_reduce_kernel<<<32, 256, 0, stream>>>(valid, ws);
    finalize_kernel<<<   1, 256, 0, stream>>>(valid, ws, out);
}